// MiniBeamformer_44375602102893
// MI455X (gfx1250) — compile-verified
//
#include <hip/hip_runtime.h>
#include <math.h>

// ---------------------------------------------------------------------------
// MiniBeamformer fully-fused CDNA5 kernel (gfx1250, wave32, WMMA f32 16x16x4)
//
// x: [B=4, 2, CH_IN=8, FREQS=256, T=2000] fp32.
// Pipeline: cLog -> cs(8->16, per-freq, WMMA) -> cAct -> fc(freq depthwise k=5)
//           -> cAct -> tc(16->8, K=80 dilated causal, WMMA) -> cExp
//           -> ls(complex 8->8, WMMA, Wr/Wi packed in M) -> cMul(x)
//           -> la(complex 8->8, WMMA) -> out
//
// Tile per block: 1 batch x 16 freqs (+2 halo each side) x 16 times (+8 hist).
// Transcendentals use native TRANS-pipe ops (co-execute with WMMA on CDNA5);
// atan2's quotient uses raw v_rcp_f32 instead of the IEEE divide sequence.
// Wave-uniform weight addresses are readfirstlane'd -> scalar loads.
// ---------------------------------------------------------------------------

#define FREQS    256
#define CH_IN    8
#define CH_INNER 16
#define TK       5
#define FK       5
#define NBATCH   4
#define NTIME    2000

#define TT    16   // output time steps per block
#define TTH   24   // + 8 causal history
#define FTILE 16   // output freqs per block
#define FH    20   // + 2 halo each side
#define NWAVE 8
#define BLOCK 256

#define TWO_PI_F     6.28318530717958647692f
#define INV_TWO_PI_F 0.15915494309189533577f
#define PI_F         3.14159265358979323846f
#define PI_2_F       1.57079632679489661923f

typedef __attribute__((ext_vector_type(2))) float v2f;
typedef __attribute__((ext_vector_type(8))) float v8f;

// D = A(16x4) * B(4x16) + C(16x16), fp32. Operand layout (wave32):
//   A: lane m=lane&15 holds row M=m; reg0 = K=2*kh+4*chunk, reg1 = K+1 (kh=lane>>4)
//   B: lane holds column N=m; reg0/reg1 same K mapping
//   C/D: VGPR j: lanes 0-15 -> (M=j, N=lane), lanes 16-31 -> (M=j+8, N=lane-16)
__device__ __forceinline__ v8f wmma4(v2f a, v2f b, v8f c) {
  return __builtin_amdgcn_wmma_f32_16x16x4_f32(false, a, false, b, (short)0, c,
                                               false, false);
}

__device__ __forceinline__ float mod2pi(float p) {
  return p - floorf(p * INV_TWO_PI_F) * TWO_PI_F;  // jnp.mod semantics
}

// Branch-free atan2 (minimax on [0,1] + quadrant fixup); ~1e-4 rad abs error.
// Quotient via raw v_rcp_f32 (TRANS pipe) -- inputs are finite, q in [0,1].
__device__ __forceinline__ float fast_atan2f(float y, float x) {
  const float ax = fabsf(x), ay = fabsf(y);
  const float mx = fmaxf(ax, ay), mn = fminf(ax, ay);
  const float a  = mn * __builtin_amdgcn_rcpf(mx);  // v_rcp_f32 + v_mul_f32
  const float ss = a * a;
  float r = fmaf(ss, -0.0464964749f, 0.15931422f);
  r = fmaf(r, ss, -0.327622764f);
  r = fmaf(a * ss, r, a);                  // ~atan(a), a in [0,1]
  r = (ay > ax) ? (PI_2_F - r) : r;
  r = (x < 0.f) ? (PI_F - r) : r;
  r = copysignf(r, y);
  r = (mx == 0.f) ? copysignf((x < 0.f) ? PI_F : 0.f, y) : r;  // atan2(0,±)
  return r;
}

__device__ __forceinline__ size_t xidx(int b, int p, int c, int f, int t) {
  return ((((size_t)b * 2 + p) * CH_IN + c) * FREQS + f) * (size_t)NTIME + t;
}

// tc GEMM for one part: D(8x16,padded to 16) = W(8x80) * Vext(80x16), K fully
// unrolled so every LDS/weight offset is an immediate.
__device__ __forceinline__ v8f tc_gemm(const float* __restrict__ tw,
                                       const float (*vb)[FTILE][TTH],
                                       int fgu, int fl, int m, int kh) {
  v8f D = {};
  const bool valid = (m < CH_IN);
  const float* wrow = tw + (size_t)(fgu * CH_IN + m) * (CH_INNER * TK);
#pragma unroll
  for (int kc = 0; kc < 20; ++kc) {
    const int k0  = kc * 4 + 2 * kh;
    const float a0 = valid ? wrow[k0]     : 0.f;
    const float a1 = valid ? wrow[k0 + 1] : 0.f;
    const int j0 = k0 / TK,       tp0 = k0 % TK;
    const int j1 = (k0 + 1) / TK, tp1 = (k0 + 1) % TK;
    const v2f Bv = {vb[j0][fl][m + 2 * tp0], vb[j1][fl][m + 2 * tp1]};
    D = wmma4((v2f){a0, a1}, Bv, D);
  }
  return D;
}

// Complex 8->8 layer (ls/la): A rows 0-7 = Wr, rows 8-15 = Wi; two WMMA chains
// against z_re / z_im staged in LDS; returns D1 (A*z_re) and D2 (A*z_im).
__device__ __forceinline__ void cplx8_gemm(const float* __restrict__ wr,
                                           const float* __restrict__ wi,
                                           const float (*zb)[CH_IN][16],
                                           int fgu, int m, int kh,
                                           v8f& D1, v8f& D2) {
  D1 = (v8f){}; D2 = (v8f){};
  const float* lw = (m < CH_IN) ? wr : wi;
  const int row = m & 7;
  const float* lrow = lw + (size_t)(fgu * CH_IN + row) * CH_IN;
#pragma unroll
  for (int kc = 0; kc < 2; ++kc) {
    const int k0 = kc * 4 + 2 * kh;
    const v2f A  = {lrow[k0], lrow[k0 + 1]};
    const v2f Br = {zb[0][k0][m], zb[0][k0 + 1][m]};
    const v2f Bi = {zb[1][k0][m], zb[1][k0 + 1][m]};
    D1 = wmma4(A, Br, D1);
    D2 = wmma4(A, Bi, D2);
  }
}

__global__ __launch_bounds__(BLOCK)
void minibf_fused_kernel(const float* __restrict__ x,
                         const float* __restrict__ cs_wr, const float* __restrict__ cs_wi,
                         const float* __restrict__ cs_br, const float* __restrict__ cs_bi,
                         const float* __restrict__ fc_wr, const float* __restrict__ fc_wi,
                         const float* __restrict__ fc_br, const float* __restrict__ fc_bi,
                         const float* __restrict__ tc_wr, const float* __restrict__ tc_wi,
                         const float* __restrict__ tc_br, const float* __restrict__ tc_bi,
                         const float* __restrict__ ls_wr, const float* __restrict__ ls_wi,
                         const float* __restrict__ ls_br, const float* __restrict__ ls_bi,
                         const float* __restrict__ la_wr, const float* __restrict__ la_wi,
                         const float* __restrict__ la_br, const float* __restrict__ la_bi,
                         float* __restrict__ out) {
  // LDS: 61440 + 49152 + 8192 = 118784 B  (2 workgroups per 320KB WGP)
  __shared__ float u[2][CH_INNER][FH][TTH];      // cAct(cs(cLog(x))) incl. freq halo
  __shared__ float vbuf[2][CH_INNER][FTILE][TTH];// cAct(fc(u))
  __shared__ float zbuf[NWAVE][2][CH_IN][16];    // per-wave re/im staging for ls/la B

  const int tid  = threadIdx.x;
  const int wave = tid >> 5;
  const int lane = tid & 31;
  const int m    = lane & 15;   // WMMA row (A) / column (B, C/D) for this lane
  const int kh   = lane >> 4;   // K-half selector within a K=4 chunk
  const int t0   = blockIdx.x * TT;
  const int f0   = blockIdx.y * FTILE;
  const int b    = blockIdx.z;

  // ---------------- Stage 1+2+3: cLog -> cs (WMMA) -> cAct -> u ----------------
  for (int fl = wave; fl < FH; fl += NWAVE) {
    const int fg = f0 + fl - 2;
    if (fg < 0 || fg >= FREQS) {
      // fc zero-pads its input along freq -> halo rows are zero.
      for (int i = lane; i < 2 * CH_INNER * TTH; i += 32) {
        const int t = i % TTH;
        const int r = (i / TTH) % CH_INNER;
        const int p = i / (TTH * CH_INNER);
        u[p][r][fl][t] = 0.f;
      }
      continue;
    }
    // fg is wave-uniform (fl stepped by wave id): force scalar addressing.
    const int fgu = __builtin_amdgcn_readfirstlane(fg);

    // A operands: per-frequency cs weights (16 out x 8 in), 2 K-chunks, 2 parts.
    float a0[2][2], a1[2][2];
#pragma unroll
    for (int p = 0; p < 2; ++p) {
      const float* W = p ? cs_wi : cs_wr;
#pragma unroll
      for (int kc = 0; kc < 2; ++kc) {
        const int k0 = kc * 4 + 2 * kh;
        a0[p][kc] = W[(fgu * CH_INNER + m) * CH_IN + k0];
        a1[p][kc] = W[(fgu * CH_INNER + m) * CH_IN + k0 + 1];
      }
    }
    // Per-row biases for this lane's two D rows (row = j + 8*kh), batched loads.
    float br_[8], bi_[8];
#pragma unroll
    for (int j = 0; j < 8; ++j) {
      br_[j] = cs_br[fgu * CH_INNER + j + 8 * kh];
      bi_[j] = cs_bi[fgu * CH_INNER + j + 8 * kh];
    }
    // Two overlapping 16-wide time subtiles cover t_local 0..23.
#pragma unroll
    for (int ts = 0; ts < 2; ++ts) {
      const int tl  = ts * 8 + m;          // this lane's t_local (B/D column)
      const int tg  = t0 - 8 + tl;
      const int tgc = tg < 0 ? 0 : tg;     // clamp; v is zeroed later for tg<0
      v2f B0[2], B1[2];                    // B built on the fly: cLog(x)
#pragma unroll
      for (int kc = 0; kc < 2; ++kc) {
        const int j0 = kc * 4 + 2 * kh;    // input channel for K row
        const float xr0 = x[xidx(b, 0, j0,     fgu, tgc)];
        const float xi0 = x[xidx(b, 1, j0,     fgu, tgc)];
        const float xr1 = x[xidx(b, 0, j0 + 1, fgu, tgc)];
        const float xi1 = x[xidx(b, 1, j0 + 1, fgu, tgc)];
        const float lm0 = 0.5f * __logf(xr0 * xr0 + xi0 * xi0 + 1e-12f);
        const float lm1 = 0.5f * __logf(xr1 * xr1 + xi1 * xi1 + 1e-12f);
        const float ph0 = fast_atan2f(xi0, xr0);
        const float ph1 = fast_atan2f(xi1, xr1);
        B0[kc] = (v2f){lm0, lm1};          // log-magnitude stream (part 0)
        B1[kc] = (v2f){ph0, ph1};          // phase stream (part 1)
      }
      v8f D0 = {}, D1 = {};
      D0 = wmma4((v2f){a0[0][0], a1[0][0]}, B0[0], D0);
      D0 = wmma4((v2f){a0[0][1], a1[0][1]}, B0[1], D0);
      D1 = wmma4((v2f){a0[1][0], a1[1][0]}, B1[0], D1);
      D1 = wmma4((v2f){a0[1][1], a1[1][1]}, B1[1], D1);
      // bias + cAct, store to u (fully unrolled: direct VGPR element access)
#pragma unroll
      for (int j = 0; j < 8; ++j) {
        const int row = j + 8 * kh;        // D VGPR j covers M=j / M=j+8
        const float re = D0[j] + br_[j];
        const float ph = D1[j] + bi_[j];
        const float w  = (__cosf(ph) + 1.f) * 0.5f;
        u[0][row][fl][tl] = w * re;
        u[1][row][fl][tl] = mod2pi(ph);
      }
    }
  }
  __syncthreads();

  // ---------------- Stage 4+5: fc (freq depthwise k=5) -> cAct -> v ----------------
  for (int pos = tid; pos < CH_INNER * FTILE * TTH; pos += BLOCK) {
    const int t  = pos % TTH;
    const int fL = (pos / TTH) % FTILE;
    const int c  = pos / (TTH * FTILE);
    const int tg = t0 - 8 + t;
    float o0 = 0.f, o1 = 0.f;
    if (tg >= 0) {                          // tc's causal zero-pad lands here
      float acc0 = fc_br[c], acc1 = fc_bi[c];
#pragma unroll
      for (int k = 0; k < FK; ++k) {        // u freq index fL+k == (fL+2)+(k-2)
        acc0 += fc_wr[c * FK + k] * u[0][c][fL + k][t];
        acc1 += fc_wi[c * FK + k] * u[1][c][fL + k][t];
      }
      const float w = (__cosf(acc1) + 1.f) * 0.5f;
      o0 = w * acc0;
      o1 = mod2pi(acc1);
    }
    vbuf[0][c][fL][t] = o0;
    vbuf[1][c][fL][t] = o1;
  }
  __syncthreads();

  // ------------- Stage 6..10: tc (WMMA K=80) -> cExp -> ls -> cMul -> la -------------
  for (int fl = wave; fl < FTILE; fl += NWAVE) {
    const int fg  = f0 + fl;
    const int fgu = __builtin_amdgcn_readfirstlane(fg);
    const float s = (kh == 0) ? -1.f : 1.f;

    // tc: out8[c][t] = sum_{j<16,k<5} w[c][j][k] * v[j][t + 2k - 8]; K = j*5+k
    const v8f E0 = tc_gemm(tc_wr, vbuf[0], fgu, fl, m, kh);
    const v8f E1 = tc_gemm(tc_wi, vbuf[1], fgu, fl, m, kh);

    // bias + cExp (valid output rows are M=0..7, i.e. lanes 0-15)
    float zr[CH_IN], zi[CH_IN];
#pragma unroll
    for (int j = 0; j < CH_IN; ++j) {
      const float rr = E0[j] + tc_br[fgu * CH_IN + j];
      const float pp = E1[j] + tc_bi[fgu * CH_IN + j];
      const float r  = __expf(rr);
      zr[j] = r * __cosf(pp);
      zi[j] = r * __sinf(pp);
    }
    if (kh == 0) {                          // lanes 0-15 hold the full 8x16 tile
#pragma unroll
      for (int j = 0; j < CH_IN; ++j) {
        zbuf[wave][0][j][m] = zr[j];
        zbuf[wave][1][j][m] = zi[j];
      }
    }
    asm volatile("s_wait_dscnt 0" ::: "memory");

    // ls: D1 = [Wr;Wi]*z_re, D2 = [Wr;Wi]*z_im
    v8f D1v, D2v;
    cplx8_gemm(ls_wr, ls_wi, zbuf[wave], fgu, m, kh, D1v, D2v);

    // yr = Wr*zr - Wi*zi (+br-bi)  [lanes 0-15];  yi = Wr*zi + Wi*zr (+br+bi) [16-31]
    float y[CH_IN];
#pragma unroll
    for (int c = 0; c < CH_IN; ++c) {
      const float o = __shfl_xor(D2v[c], 16, 32);
      y[c] = D1v[c] + s * o + ls_br[fgu * CH_IN + c] + s * ls_bi[fgu * CH_IN + c];
    }
    // cMul with original x (re-read; whole x is L2-resident)
    float z2[CH_IN];
#pragma unroll
    for (int c = 0; c < CH_IN; ++c) {
      const float xr  = x[xidx(b, 0, c, fgu, t0 + m)];
      const float xi  = x[xidx(b, 1, c, fgu, t0 + m)];
      const float par = __shfl_xor(y[c], 16, 32);
      z2[c] = y[c] * xr + s * par * xi;     // re: yr*xr - yi*xi ; im: yi*xr + yr*xi
    }
#pragma unroll
    for (int c = 0; c < CH_IN; ++c)         // kh==0 lanes: re plane; kh==1: im plane
      zbuf[wave][kh][c][m] = z2[c];
    asm volatile("s_wait_dscnt 0" ::: "memory");

    // la: same complex 8->8 pattern
    v8f G1, G2;
    cplx8_gemm(la_wr, la_wi, zbuf[wave], fgu, m, kh, G1, G2);
#pragma unroll
    for (int c = 0; c < CH_IN; ++c) {
      const float o = __shfl_xor(G2[c], 16, 32);
      const float r = G1[c] + s * o + la_br[fgu * CH_IN + c] + s * la_bi[fgu * CH_IN + c];
      out[xidx(b, kh, c, fgu, t0 + m)] = r;  // lanes 0-15 -> re plane, 16-31 -> im
    }
  }
}

extern "C" void kernel_launch(void* const* d_in, const int* in_sizes, int n_in,
                              void* d_out, int out_size, void* d_ws, size_t ws_size,
                              hipStream_t stream) {
  (void)in_sizes; (void)n_in; (void)d_ws; (void)ws_size; (void)out_size;
  const float* x     = (const float*)d_in[0];
  const float* cs_wr = (const float*)d_in[1];
  const float* cs_wi = (const float*)d_in[2];
  const float* cs_br = (const float*)d_in[3];
  const float* cs_bi = (const float*)d_in[4];
  const float* fc_wr = (const float*)d_in[5];
  const float* fc_wi = (const float*)d_in[6];
  const float* fc_br = (const float*)d_in[7];
  const float* fc_bi = (const float*)d_in[8];
  const float* tc_wr = (const float*)d_in[9];
  const float* tc_wi = (const float*)d_in[10];
  const float* tc_br = (const float*)d_in[11];
  const float* tc_bi = (const float*)d_in[12];
  const float* ls_wr = (const float*)d_in[13];
  const float* ls_wi = (const float*)d_in[14];
  const float* ls_br = (const float*)d_in[15];
  const float* ls_bi = (const float*)d_in[16];
  const float* la_wr = (const float*)d_in[17];
  const float* la_wi = (const float*)d_in[18];
  const float* la_br = (const float*)d_in[19];
  const float* la_bi = (const float*)d_in[20];

  dim3 grid(NTIME / TT, FREQS / FTILE, NBATCH);   // 125 x 16 x 4
  dim3 block(BLOCK);
  minibf_fused_kernel<<<grid, block, 0, stream>>>(
      x, cs_wr, cs_wi, cs_br, cs_bi, fc_wr, fc_wi, fc_br, fc_bi,
      tc_wr, tc_wi, tc_br, tc_bi, ls_wr, ls_wi, ls_br, ls_bi,
      la_wr, la_wi, la_br, la_bi, (float*)d_out);
}